// attention_35673998360872
// MI455X (gfx1250) — compile-verified
//
#include <hip/hip_runtime.h>

typedef float v2f __attribute__((ext_vector_type(2)));
typedef float v8f __attribute__((ext_vector_type(8)));
typedef unsigned int v4u __attribute__((ext_vector_type(4)));
typedef int v8i __attribute__((ext_vector_type(8)));
typedef int v4i __attribute__((ext_vector_type(4)));

#define B_   64
#define K_   256
#define T_   64
#define DF_  768
#define N_   1024
#define NT   64     // n-tile width
#define DC   64     // d-chunk depth for GEMM1 staging
#define NCH  (DF_/DC)
#define EP   66     // e LDS pitch  (64 + pad 2 dwords)
#define VP   65     // v LDS pitch
#define FP   68     // f-chunk LDS pitch (64 + pad 4 dwords)
#define SP   65     // s LDS pitch

// e 256*66 + v 256*65 + 2*f 64*68 + s 64*65 + 4*64 + 256 bias
#define LDS_FLOATS (K_*EP + K_*VP + 2*DC*FP + T_*SP + 4*T_ + K_)

static __device__ __forceinline__ v8f wmma4(v2f a, v2f b, v8f c) {
  // D = A(16x4 f32) * B(4x16 f32) + C(16x16 f32)
  return __builtin_amdgcn_wmma_f32_16x16x4_f32(false, a, false, b, (short)0, c,
                                               false, false);
}

// raw LDS byte offset of a generic pointer into shared memory
static __device__ __forceinline__ unsigned lds_offset_of(const void* p) {
  return (unsigned)(size_t)(__attribute__((address_space(3))) const void*)p;
}

// TDM: async-load a 2D fp32 tile (tile_d0 x tile_d1) from global into LDS,
// with per-row LDS padding (pitch = row + pad). Tracked by TENSORcnt.
static __device__ __forceinline__ void tdm_load_2d(const void* gaddr,
                                                   unsigned lds_off,
                                                   unsigned tensor_d0,
                                                   unsigned tensor_d1,
                                                   unsigned stride0,
                                                   unsigned tile_d0,
                                                   unsigned tile_d1,
                                                   unsigned pad_interval_code,
                                                   unsigned pad_amount_code) {
  unsigned long long ga = (unsigned long long)(size_t)gaddr;
  v4u g0;
  g0.x = 1u;                                   // count=1, user descriptor
  g0.y = lds_off;                              // lds_addr (bytes)
  g0.z = (unsigned)ga;                         // global_addr[31:0]
  g0.w = (unsigned)((ga >> 32) & 0x01FFFFFFu)  // global_addr[56:32]
         | (2u << 30);                         // type = 2 ("image")
  v8i g1;
  unsigned w0 = (2u << 16)                     // data_size = 4 bytes
              | (1u << 20)                     // pad_enable
              | (pad_interval_code << 22)      // pad every 2<<code DWORDs
              | (pad_amount_code << 25);       // pad (code+1) DWORDs
  g1[0] = (int)w0;                             // workgroup_mask=0 (no cluster)
  g1[1] = (int)((tensor_d0 & 0xFFFFu) << 16);  // abar_addr=0 | tensor_dim0 lo
  g1[2] = (int)(((tensor_d0 >> 16) & 0xFFFFu) | ((tensor_d1 & 0xFFFFu) << 16));
  g1[3] = (int)(((tensor_d1 >> 16) & 0xFFFFu) | ((tile_d0 & 0xFFFFu) << 16));
  g1[4] = (int)(tile_d1 & 0xFFFFu);            // tile_dim1 | tile_dim2=0 (2D)
  g1[5] = (int)stride0;                        // tensor_dim0_stride[31:0]
  g1[6] = 0;                                   // stride0 hi | stride1 lo
  g1[7] = 0;                                   // stride1 hi
  v4i gz4 = {0, 0, 0, 0};                      // groups 2/3 unused (<=2D)
  v8i gz8 = {0, 0, 0, 0, 0, 0, 0, 0};          // extra group (unused)
  __builtin_amdgcn_tensor_load_to_lds(g0, g1, gz4, gz4, gz8, 0);
}

__global__ __launch_bounds__(256, 1)
void damsm_fused_kernel(const float* __restrict__ eg,   // [B,K,T]
                        const float* __restrict__ fg,   // [B,DF,N]
                        const float* __restrict__ gptr, // [1]
                        const float* __restrict__ Wg,   // [DF,K]
                        const float* __restrict__ bg,   // [K]
                        float* __restrict__ out) {      // [B,T,T]
  __shared__ float lds[LDS_FLOATS];
  float* e_s  = lds;                 // K_*EP
  float* v_s  = e_s + K_*EP;         // K_*VP
  float* f_s0 = v_s + K_*VP;         // DC*FP  (double-buffered f chunk)
  float* f_s1 = f_s0 + DC*FP;        // DC*FP
  float* s_s  = f_s1 + DC*FP;        // T_*SP
  float* m_s  = s_s + T_*SP;         // T_
  float* sc_s = m_s + T_;            // T_
  float* lc_s = sc_s + T_;           // T_
  float* le_s = lc_s + T_;           // T_
  float* b_s  = le_s + T_;           // K_

  const int tid  = threadIdx.x;
  const int lane = tid & 31;
  const int wv   = tid >> 5;       // 0..7
  const int ml   = lane & 15;      // M (or N) index within a fragment
  const int hi   = lane >> 4;      // half-wave select
  const int dkb  = hi * 2;         // inner-dim base for A/B fragments
  const int h8   = hi * 8;         // M offset for C/D fragments
  const int b    = blockIdx.x;
  const float gamma = gptr[0];

  const float* eb = eg + (size_t)b * K_ * T_;
  const float* fb = fg + (size_t)b * DF_ * N_;
  const unsigned foff[2] = {lds_offset_of(f_s0), lds_offset_of(f_s1)};

  // ---- TDM: stage e[b] (256x64 -> pitch 66: pad 2 dwords per 64) ----
  if (wv == 0)
    tdm_load_2d(eb, lds_offset_of(e_s), T_, K_, T_, T_, K_, /*64dw*/5, /*2dw*/1);
  if (tid < K_) b_s[tid] = bg[tid];
  if (tid < T_) m_s[tid] = -__builtin_inff();

  const v8f vzero = {0.f, 0.f, 0.f, 0.f, 0.f, 0.f, 0.f, 0.f};
  v8f cacc[2][4];  // cAcc rows [32*wv, 32*wv+32) x all 64 t-cols, in VGPRs
  #pragma unroll
  for (int rt = 0; rt < 2; ++rt)
    #pragma unroll
    for (int ct = 0; ct < 4; ++ct) cacc[rt][ct] = vzero;

  for (int n0 = 0; n0 < N_; n0 += NT) {
    // ================= Phase A: v_tile = W^T * f_tile + bias =================
    v8f vacc[2][4];
    #pragma unroll
    for (int rt = 0; rt < 2; ++rt)
      #pragma unroll
      for (int ct = 0; ct < 4; ++ct) vacc[rt][ct] = vzero;

    // prologue: async-stage chunk 0 of this n-tile
    if (wv == 0)
      tdm_load_2d(fb + n0, foff[0], N_, DF_, N_, NT, DC, /*64dw*/5, /*4dw*/3);

    for (int ci = 0; ci < NCH; ++ci) {
      const int d0 = ci * DC;
      if (wv == 0) {
        if (ci + 1 < NCH) {
          // overlap: stage next chunk into the other buffer, keep 1 in flight
          tdm_load_2d(fb + (size_t)(d0 + DC) * N_ + n0, foff[(ci + 1) & 1],
                      N_, DF_, N_, NT, DC, 5, 3);
          __builtin_amdgcn_s_wait_tensorcnt(1);
        } else {
          __builtin_amdgcn_s_wait_tensorcnt(0);
        }
      }
      __syncthreads();  // chunk ci staged; prev-phase LDS consumers done
      const float* fcur = (ci & 1) ? f_s1 : f_s0;

      for (int ds = 0; ds < DC; ds += 4) {
        const float* wrow = Wg + (size_t)(d0 + ds + dkb) * K_ + 32 * wv + ml;
        v2f a0, a1;
        a0.x = wrow[0];        a0.y = wrow[K_];        // rt=0: M rows 32w..+15
        a1.x = wrow[16];       a1.y = wrow[K_ + 16];   // rt=1: M rows +16..+31
        v2f bf[4];
        #pragma unroll
        for (int ct = 0; ct < 4; ++ct) {
          bf[ct].x = fcur[(ds + dkb) * FP + ct * 16 + ml];
          bf[ct].y = fcur[(ds + dkb + 1) * FP + ct * 16 + ml];
        }
        #pragma unroll
        for (int ct = 0; ct < 4; ++ct) {
          vacc[0][ct] = wmma4(a0, bf[ct], vacc[0][ct]);
          vacc[1][ct] = wmma4(a1, bf[ct], vacc[1][ct]);
        }
      }
      __syncthreads();  // all waves done with buffer (ci&1) before re-staging
    }

    // write v_tile (+bias) to LDS
    #pragma unroll
    for (int rt = 0; rt < 2; ++rt) {
      int k0 = 32 * wv + 16 * rt + h8;
      #pragma unroll
      for (int ct = 0; ct < 4; ++ct)
        #pragma unroll
        for (int r = 0; r < 8; ++r)
          v_s[(k0 + r) * VP + ct * 16 + ml] = vacc[rt][ct][r] + b_s[k0 + r];
    }
    __syncthreads();

    // ================= Phase B: s = gamma * e^T v_tile =================
    {
      int t0 = 16 * (wv >> 1);
      int nh = wv & 1;
      v8f sacc[2] = {vzero, vzero};
      for (int kk = 0; kk < K_; kk += 4) {
        v2f a;
        a.x = e_s[(kk + dkb) * EP + t0 + ml];
        a.y = e_s[(kk + dkb + 1) * EP + t0 + ml];
        #pragma unroll
        for (int c = 0; c < 2; ++c) {
          int nn0 = 16 * (2 * nh + c);
          v2f bf;
          bf.x = v_s[(kk + dkb) * VP + nn0 + ml];
          bf.y = v_s[(kk + dkb + 1) * VP + nn0 + ml];
          sacc[c] = wmma4(a, bf, sacc[c]);
        }
      }
      #pragma unroll
      for (int c = 0; c < 2; ++c) {
        int nn0 = 16 * (2 * nh + c);
        #pragma unroll
        for (int r = 0; r < 8; ++r)
          s_s[(t0 + r + h8) * SP + nn0 + ml] = gamma * sacc[c][r];
      }
    }
    __syncthreads();

    // ========== Phase C: online max + exp (one token row per thread) ==========
    if (tid < T_) {
      float* row = s_s + tid * SP;
      float tmax = row[0];
      for (int n = 1; n < NT; ++n) tmax = fmaxf(tmax, row[n]);
      float mo = m_s[tid];
      float mn = fmaxf(mo, tmax);
      sc_s[tid] = __expf(mo - mn);   // 0 on first tile (mo = -inf)
      m_s[tid]  = mn;
      for (int n = 0; n < NT; ++n) row[n] = __expf(row[n] - mn);
    }
    __syncthreads();

    // ========== Phase D: rescale running c-accumulators (per t column) ==========
    #pragma unroll
    for (int ct = 0; ct < 4; ++ct) {
      float fsc = sc_s[ct * 16 + ml];
      #pragma unroll
      for (int rt = 0; rt < 2; ++rt)
        #pragma unroll
        for (int r = 0; r < 8; ++r) cacc[rt][ct][r] *= fsc;
    }

    // ================= Phase E: cAcc += v_tile * p^T =================
    for (int nn = 0; nn < NT; nn += 4) {
      v2f a0, a1;
      a0.x = v_s[(32 * wv + ml) * VP + nn + dkb];
      a0.y = v_s[(32 * wv + ml) * VP + nn + dkb + 1];
      a1.x = v_s[(32 * wv + 16 + ml) * VP + nn + dkb];
      a1.y = v_s[(32 * wv + 16 + ml) * VP + nn + dkb + 1];
      v2f bf[4];
      #pragma unroll
      for (int ct = 0; ct < 4; ++ct) {
        bf[ct].x = s_s[(ct * 16 + ml) * SP + nn + dkb];
        bf[ct].y = s_s[(ct * 16 + ml) * SP + nn + dkb + 1];
      }
      #pragma unroll
      for (int ct = 0; ct < 4; ++ct) {
        cacc[0][ct] = wmma4(a0, bf[ct], cacc[0][ct]);
        cacc[1][ct] = wmma4(a1, bf[ct], cacc[1][ct]);
      }
    }
    __syncthreads();
  }  // n-tile loop

  // ---- spill unnormalized c-hat to LDS (v_s reused); softmax denom cancels ----
  #pragma unroll
  for (int rt = 0; rt < 2; ++rt) {
    int k0 = 32 * wv + 16 * rt + h8;
    #pragma unroll
    for (int ct = 0; ct < 4; ++ct)
      #pragma unroll
      for (int r = 0; r < 8; ++r)
        v_s[(k0 + r) * VP + ct * 16 + ml] = cacc[rt][ct][r];
  }
  __syncthreads();

  // ---- per-token norms over K ----
  if (tid < T_) {
    float sc2 = 0.f, se2 = 0.f;
    for (int k = 0; k < K_; ++k) {
      float cv = v_s[k * VP + tid];
      float ev = e_s[k * EP + tid];
      sc2 += cv * cv;
      se2 += ev * ev;
    }
    lc_s[tid] = sqrtf(sc2);
    le_s[tid] = sqrtf(se2);
  }
  __syncthreads();

  // ============ Phase F: cos = (c-hat^T e) / (||c-hat|| ||e||) ============
  {
    int t0 = 16 * (wv >> 1);  // output row tile i
    int nh = wv & 1;
    v8f oacc[2] = {vzero, vzero};
    for (int kk = 0; kk < K_; kk += 4) {
      v2f a;
      a.x = v_s[(kk + dkb) * VP + t0 + ml];
      a.y = v_s[(kk + dkb + 1) * VP + t0 + ml];
      #pragma unroll
      for (int c = 0; c < 2; ++c) {
        int j0 = 16 * (2 * nh + c);
        v2f bf;
        bf.x = e_s[(kk + dkb) * EP + j0 + ml];
        bf.y = e_s[(kk + dkb + 1) * EP + j0 + ml];
        oacc[c] = wmma4(a, bf, oacc[c]);
      }
    }
    float* ob = out + (size_t)b * T_ * T_;
    #pragma unroll
    for (int c = 0; c < 2; ++c) {
      int j = 16 * (2 * nh + c) + ml;
      float invle = 1.0f / le_s[j];
      #pragma unroll
      for (int r = 0; r < 8; ++r) {
        int i = t0 + r + h8;
        ob[i * T_ + j] = oacc[c][r] * invle / lc_s[i];
      }
    }
  }
}

extern "C" void kernel_launch(void* const* d_in, const int* in_sizes, int n_in,
                              void* d_out, int out_size, void* d_ws, size_t ws_size,
                              hipStream_t stream) {
  const float* e     = (const float*)d_in[0];
  const float* f     = (const float*)d_in[1];
  const float* gamma = (const float*)d_in[2];
  const float* W     = (const float*)d_in[3];
  const float* bias  = (const float*)d_in[4];
  float* out = (float*)d_out;
  (void)in_sizes; (void)n_in; (void)out_size; (void)d_ws; (void)ws_size;
  damsm_fused_kernel<<<dim3(B_), dim3(256), 0, stream>>>(e, f, gamma, W, bias, out);
}